// VectorQuantizer_2DCB_35639638622553
// MI455X (gfx1250) — compile-verified
//
#include <hip/hip_runtime.h>

// CDNA5 / gfx1250 VQ kernel: fp32 WMMA 16x16x4 scores, LDS-staged codebook.

typedef float v2f __attribute__((ext_vector_type(2)));
typedef float v8f __attribute__((ext_vector_type(8)));

#define N_E        1024
#define E_DIM      64
#define LDS_STRIDE 68            // 64 + 4 pad -> conflict-free 16-lane b64 reads
#define NUM_WG     256
#define WAVES_WG   8
#define TILES_WAVE 2             // 256*8*2 = 4096 tiles of 16 rows = 65536 rows
#define Z_ELEMS    4194304       // 16*64*64*64

__global__ __launch_bounds__(256) void vq_main(const float* __restrict__ z,
                                               const float* __restrict__ cbk,
                                               float* __restrict__ out,
                                               float* __restrict__ partial) {
    extern __shared__ float smem[];
    float* Elds = smem;                       // N_E * LDS_STRIDE
    float* ssq  = smem + N_E * LDS_STRIDE;    // N_E
    float* wsum = ssq + N_E;                  // WAVES_WG

    const int t = threadIdx.x;

    // ---- stage codebook into LDS (coalesced float4), sumsq via ds_add_f32 ----
    for (int r = t; r < N_E; r += 256) ssq[r] = 0.0f;
    __syncthreads();
    for (int i = t; i < N_E * 16; i += 256) {
        int r = i >> 4;
        int j = (i & 15) << 2;                 // float offset within row
        float4 v = ((const float4*)cbk)[i];    // cbk + r*64 + j
        *(float4*)&Elds[r * LDS_STRIDE + j] = v;
        atomicAdd(&ssq[r], v.x * v.x + v.y * v.y + v.z * v.z + v.w * v.w);
    }
    __syncthreads();

    const int lane = t & 31;
    const int wave = t >> 5;
    const int row  = lane & 15;   // M index (mod 8 for C layout) / column-in-block
    const int hi   = lane >> 4;   // K sub-pair selector for A/B fragments

    float lsum = 0.0f;

    for (int tt = 0; tt < TILES_WAVE; ++tt) {
        const int tile = blockIdx.x * WAVES_WG + wave + tt * (NUM_WG * WAVES_WG);
        const int n0 = tile << 4;
        const int bb = n0 >> 12;
        const int hh = (n0 >> 6) & 63;
        const int w0 = n0 & 63;
        const long base = ((long)bb << 18) + (hh << 6) + w0 + row;
        const float* zb = z + base;            // element (row, c) at zb[c*4096]

        // ---- load A fragments: Z[row, k], lanes 0-15 K={0,1}, 16-31 K={2,3} ----
        v2f a[16];
#pragma unroll
        for (int kb = 0; kb < 16; ++kb) {
            int c = (kb << 2) + (hi << 1);
            a[kb].x = zb[(long)c * 4096];
            a[kb].y = zb[(long)(c + 1) * 4096];
        }

        float bestd[8];
        int   besti[8];
#pragma unroll
        for (int r = 0; r < 8; ++r) { bestd[r] = 3.4e38f; besti[r] = 0; }

        // ---- 64 codebook column blocks: 16 WMMA (K=64) each, then argmin ----
        for (int cb = 0; cb < 64; ++cb) {
            const int col = (cb << 4) + row;
            const float* ebase = &Elds[col * LDS_STRIDE + (hi << 1)];
            v8f acc = {};
#pragma unroll
            for (int kb = 0; kb < 16; ++kb) {
                v2f bf = *(const v2f*)(ebase + (kb << 2));
                acc = __builtin_amdgcn_wmma_f32_16x16x4_f32(
                    false, a[kb], false, bf, (short)0, acc, false, false);
            }
            const float ss = ssq[col];
#pragma unroll
            for (int r = 0; r < 8; ++r) {
                float d = __builtin_fmaf(-2.0f, acc[r], ss);
                bool take = d < bestd[r];
                bestd[r] = take ? d : bestd[r];
                besti[r] = take ? col : besti[r];
            }
        }

        // ---- cross-lane argmin within each 16-lane half (rows 0-7 / 8-15) ----
#pragma unroll
        for (int m = 1; m <= 8; m <<= 1) {
#pragma unroll
            for (int r = 0; r < 8; ++r) {
                float od = __shfl_xor(bestd[r], m, 32);
                int   oi = __shfl_xor(besti[r], m, 32);
                bool take = (od < bestd[r]) || ((od == bestd[r]) && (oi < besti[r]));
                bestd[r] = take ? od : bestd[r];
                besti[r] = take ? oi : besti[r];
            }
        }

        // ---- fetch this lane's row index: besti[row&7] from half (row>>3) ----
        const int rs = row & 7;
        int sel = besti[0];
        sel = (rs == 1) ? besti[1] : sel;
        sel = (rs == 2) ? besti[2] : sel;
        sel = (rs == 3) ? besti[3] : sel;
        sel = (rs == 4) ? besti[4] : sel;
        sel = (rs == 5) ? besti[5] : sel;
        sel = (rs == 6) ? besti[6] : sel;
        sel = (rs == 7) ? besti[7] : sel;
        const int src = ((row & 8) << 1) | (row & 7);
        const int myidx = __shfl(sel, src, 32);

        // ---- gather z_q from LDS, write (b,c,h,w) output, accumulate loss ----
        const float* eb = &Elds[myidx * LDS_STRIDE];
        float* ob = out + base;
#pragma unroll
        for (int kb = 0; kb < 16; ++kb) {
            int c = (kb << 2) + (hi << 1);
            v2f e = *(const v2f*)(eb + c);
            float d0 = e.x - a[kb].x;
            float d1 = e.y - a[kb].y;
            lsum += d0 * d0 + d1 * d1;
            ob[(long)c * 4096]       = e.x;
            ob[(long)(c + 1) * 4096] = e.y;
        }
    }

    // ---- loss: wave reduce, then workgroup partial to d_ws ----
#pragma unroll
    for (int m = 1; m < 32; m <<= 1) lsum += __shfl_xor(lsum, m, 32);
    if (lane == 0) wsum[wave] = lsum;
    __syncthreads();
    if (t == 0) {
        float s = 0.0f;
#pragma unroll
        for (int i = 0; i < WAVES_WG; ++i) s += wsum[i];
        partial[blockIdx.x] = s;
    }
}

__global__ __launch_bounds__(256) void vq_reduce(const float* __restrict__ partial,
                                                 float* __restrict__ loss_out) {
    __shared__ float sh[256];
    sh[threadIdx.x] = partial[threadIdx.x];   // NUM_WG == 256
    __syncthreads();
    for (int st = 128; st > 0; st >>= 1) {
        if (threadIdx.x < st) sh[threadIdx.x] += sh[threadIdx.x + st];
        __syncthreads();
    }
    if (threadIdx.x == 0)
        loss_out[0] = 1.25f * sh[0] / (float)Z_ELEMS;  // (1 + BETA) * mean
}

extern "C" void kernel_launch(void* const* d_in, const int* in_sizes, int n_in,
                              void* d_out, int out_size, void* d_ws, size_t ws_size,
                              hipStream_t stream) {
    const float* z   = (const float*)d_in[0];   // (16,64,64,64) fp32
    const float* cbk = (const float*)d_in[1];   // (1024,64) fp32
    float* out = (float*)d_out;                 // [z_q flat | loss]
    float* partial = (float*)d_ws;              // NUM_WG floats

    const size_t smem = (size_t)(N_E * LDS_STRIDE + N_E + WAVES_WG) * sizeof(float);
    vq_main<<<NUM_WG, 256, smem, stream>>>(z, cbk, out, partial);
    vq_reduce<<<1, 256, 0, stream>>>(partial, out + Z_ELEMS);
}